// KeypointNetwithIOLoss_43920335569252
// MI455X (gfx1250) — compile-verified
//
#include <hip/hip_runtime.h>
#include <math.h>

typedef float v2f __attribute__((ext_vector_type(2)));
typedef float v8f __attribute__((ext_vector_type(8)));

#define B_ 4
#define C_ 256
#define H_ 60
#define W_ 80
#define N_ (H_*W_)
#define EPSF 1e-8f
#define PD_EPSF 1e-6f
#define MARGINF 0.2f

// ---------------------------------------------------------------------------
// Kernel 1: bilinear sample (align_corners=True) + column L2-normalize.
// One block (256 threads == C) per sampled point; grid covers ref then tar.
// Output layout (B, N, C) so that WMMA A/B fragments are contiguous float2s.
// ---------------------------------------------------------------------------
__global__ void __launch_bounds__(256) sample_norm_kernel(
    const float* __restrict__ srcDes, const float* __restrict__ tgtDes,
    const float* __restrict__ srcPts, const float* __restrict__ tgtPts,
    float* __restrict__ refF, float* __restrict__ tarF) {
  int blk = blockIdx.x;
  int which = blk / (B_ * N_);            // 0 = ref, 1 = tar
  int p = blk - which * (B_ * N_);
  int b = p / N_;
  int i = p - b * N_;

  const float* des = which ? tgtDes : srcDes;
  const float* pts = which ? tgtPts : srcPts;
  float* out = which ? tarF : refF;

  float gx = pts[((size_t)b * N_ + i) * 2 + 0];
  float gy = pts[((size_t)b * N_ + i) * 2 + 1];
  float x = (gx + 1.0f) * ((float)(W_ - 1) * 0.5f);
  float y = (gy + 1.0f) * ((float)(H_ - 1) * 0.5f);
  float x0f = fminf(fmaxf(floorf(x), 0.0f), (float)(W_ - 1));
  float y0f = fminf(fmaxf(floorf(y), 0.0f), (float)(H_ - 1));
  float x1f = fminf(x0f + 1.0f, (float)(W_ - 1));
  float y1f = fminf(y0f + 1.0f, (float)(H_ - 1));
  float wx = x - x0f, wy = y - y0f;
  int x0 = (int)x0f, x1 = (int)x1f, y0 = (int)y0f, y1 = (int)y1f;

  int c = threadIdx.x;
  const float* img = des + ((size_t)b * C_ + c) * (size_t)(H_ * W_);
  float v00 = img[y0 * W_ + x0];
  float v01 = img[y0 * W_ + x1];
  float v10 = img[y1 * W_ + x0];
  float v11 = img[y1 * W_ + x1];
  float v = v00 * (1.0f - wx) * (1.0f - wy) + v01 * wx * (1.0f - wy) +
            v10 * (1.0f - wx) * wy + v11 * wx * wy;

  __shared__ float red[256];
  float t = v + EPSF;                      // norm(ref + EPS)
  red[c] = t * t;
  __syncthreads();
  for (int s = 128; s > 0; s >>= 1) {
    if (c < s) red[c] += red[c + s];
    __syncthreads();
  }
  float denom = sqrtf(red[0]) + EPSF;      // ... + EPS
  out[((size_t)b * N_ + i) * C_ + c] = v / denom;   // numerator is raw v
}

// ---------------------------------------------------------------------------
// Kernel 2: fused  dmat = ref.T @ tar  (V_WMMA_F32_16X16X4_F32, K-loop of 64)
// + per-row argmin (recall NN) + hardest-negative argmin, never materializing
// the 4800x4800 matrix. One block = 16 query rows; 8 waves stripe the 300
// candidate 16-tiles. A-tile (16x256 f32) staged in LDS (stride 260 => no
// bank conflicts for b64 reads).
// ---------------------------------------------------------------------------
__global__ void __launch_bounds__(256) gemm_argmin_kernel(
    const float* __restrict__ refF, const float* __restrict__ tarF,
    const float* __restrict__ tarUn, const int* __restrict__ relaxPtr,
    int* __restrict__ mjOut, int* __restrict__ njOut) {
  const int blk = blockIdx.x;
  const int b = blk / (N_ / 16);
  const int rb = blk - b * (N_ / 16);
  const int row0 = rb * 16;

  __shared__ float sA[16][260];
  __shared__ float wmd[8][16];
  __shared__ int   wmj[8][16];
  __shared__ float wnd[8][16];
  __shared__ int   wnj[8][16];

  const float* refBase = refF + ((size_t)b * N_ + row0) * C_;
  for (int idx = threadIdx.x; idx < 16 * C_; idx += 256) {
    int r = idx >> 8;
    int cc = idx & 255;
    sA[r][cc] = refBase[r * C_ + cc];
  }
  __syncthreads();

  const int wave = threadIdx.x >> 5;
  const int lane = threadIdx.x & 31;
  const int g = lane >> 4;        // lane half selects K pair {2g, 2g+1}
  const int ln = lane & 15;       // M for A-frag, N for B-frag, N for C/D

  const float rff = (float)relaxPtr[0];
  const float* tunx = tarUn + (size_t)b * 2 * N_;
  const float* tuny = tunx + N_;

  // True coords for this lane's 8 output rows (m = r + 8*g)
  float ixr[8], iyr[8];
  for (int r = 0; r < 8; ++r) {
    int m = r + 8 * g;
    ixr[r] = tunx[row0 + m];
    iyr[r] = tuny[row0 + m];
  }

  float md[8], nd[8];
  int mj[8], nj[8];
  for (int r = 0; r < 8; ++r) { md[r] = 3.0e38f; nd[r] = 3.0e38f; mj[r] = 0; nj[r] = 0; }

  const float* arow = &sA[ln][2 * g];
  for (int cb = wave; cb < N_ / 16; cb += 8) {
    int col = cb * 16;
    const float* bbase = tarF + ((size_t)b * N_ + col + ln) * C_ + 2 * g;
    v8f acc = {0.f, 0.f, 0.f, 0.f, 0.f, 0.f, 0.f, 0.f};
    for (int k0 = 0; k0 < C_; k0 += 4) {
      v2f av = *(const v2f*)(arow + k0);
      v2f bv = *(const v2f*)(bbase + k0);
      acc = __builtin_amdgcn_wmma_f32_16x16x4_f32(
          false, av, false, bv, (short)0, acc, false, false);
    }
    int j = col + ln;
    float jx = tunx[j], jy = tuny[j];
    for (int r = 0; r < 8; ++r) {
      float cl = fminf(fmaxf(acc[r], -1.0f), 1.0f);
      float d = sqrtf(2.0f - 2.0f * cl + EPSF);
      if (d < md[r] || (d == md[r] && j < mj[r])) { md[r] = d; mj[r] = j; }
      bool corr = (fabsf(jx - ixr[r]) <= rff) && (fabsf(jy - iyr[r]) <= rff);
      if (!corr) {
        if (d < nd[r] || (d == nd[r] && j < nj[r])) { nd[r] = d; nj[r] = j; }
      }
    }
  }

  // Reduce over the 16 lanes of each half (N dimension of the tile)
  for (int mask = 1; mask < 16; mask <<= 1) {
    for (int r = 0; r < 8; ++r) {
      float od = __shfl_xor(md[r], mask, 32);
      int oj = __shfl_xor(mj[r], mask, 32);
      if (od < md[r] || (od == md[r] && oj < mj[r])) { md[r] = od; mj[r] = oj; }
      float pd = __shfl_xor(nd[r], mask, 32);
      int pj = __shfl_xor(nj[r], mask, 32);
      if (pd < nd[r] || (pd == nd[r] && pj < nj[r])) { nd[r] = pd; nj[r] = pj; }
    }
  }
  if (ln == 0) {
    for (int r = 0; r < 8; ++r) {
      int m = r + 8 * g;
      wmd[wave][m] = md[r]; wmj[wave][m] = mj[r];
      wnd[wave][m] = nd[r]; wnj[wave][m] = nj[r];
    }
  }
  __syncthreads();

  if (threadIdx.x < 16) {
    int m = threadIdx.x;
    float MD = wmd[0][m]; int MJ = wmj[0][m];
    float ND = wnd[0][m]; int NJ = wnj[0][m];
    for (int w = 1; w < 8; ++w) {
      float od = wmd[w][m]; int oj = wmj[w][m];
      if (od < MD || (od == MD && oj < MJ)) { MD = od; MJ = oj; }
      float pd = wnd[w][m]; int pj = wnj[w][m];
      if (pd < ND || (pd == ND && pj < NJ)) { ND = pd; NJ = pj; }
    }
    // If no incorrect candidate beats the 2.0 sentinel, reference's argmin
    // lands on sorted rank 0 -> candidates[0] (overall argmin).
    int neg = (ND >= 2.0f && MD < ND) ? MJ : NJ;
    int i = row0 + m;
    mjOut[b * N_ + i] = MJ;
    njOut[b * N_ + i] = neg;
  }
}

// ---------------------------------------------------------------------------
// Kernel 3: per-query triplet loss + recall indicator. 16 lanes per query.
// ---------------------------------------------------------------------------
__global__ void __launch_bounds__(256) loss_kernel(
    const float* __restrict__ refF, const float* __restrict__ tarF,
    const float* __restrict__ tarUn, const int* __restrict__ mjArr,
    const int* __restrict__ njArr, float* __restrict__ lossArr,
    float* __restrict__ recArr) {
  int gi = blockIdx.x * 16 + (threadIdx.x >> 4);
  int cl = threadIdx.x & 15;
  int b = gi / N_;
  int ii = gi - b * N_;

  const float* refRow = refF + (size_t)gi * C_;
  const float* tarRow = tarF + (size_t)gi * C_;
  int nj = njArr[gi];
  const float* negRow = tarF + ((size_t)b * N_ + nj) * C_;

  float sp = 0.0f, sn = 0.0f;
  for (int c = cl; c < C_; c += 16) {
    float dp = refRow[c] - tarRow[c] + PD_EPSF;
    sp += dp * dp;
    float dn = refRow[c] - negRow[c] + PD_EPSF;
    sn += dn * dn;
  }
  for (int mask = 1; mask < 16; mask <<= 1) {
    sp += __shfl_xor(sp, mask, 32);
    sn += __shfl_xor(sn, mask, 32);
  }
  if (cl == 0) {
    lossArr[gi] = fmaxf(sqrtf(sp) - sqrtf(sn) + MARGINF, 0.0f);
    const float* tunx = tarUn + (size_t)b * 2 * N_;
    const float* tuny = tunx + N_;
    int mj = mjArr[gi];
    float rec = (fabsf(tunx[mj] - tunx[ii]) == 0.0f &&
                 fabsf(tuny[mj] - tuny[ii]) == 0.0f) ? 1.0f : 0.0f;
    recArr[gi] = rec;
  }
}

// ---------------------------------------------------------------------------
// Kernel 4: deterministic single-block reduction -> (loss_mean, recall_mean)
// ---------------------------------------------------------------------------
__global__ void __launch_bounds__(256) reduce_kernel(
    const float* __restrict__ lossArr, const float* __restrict__ recArr,
    float* __restrict__ out) {
  __shared__ float sl[256];
  __shared__ float sr[256];
  int t = threadIdx.x;
  float a = 0.0f, r = 0.0f;
  for (int i = t; i < B_ * N_; i += 256) { a += lossArr[i]; r += recArr[i]; }
  sl[t] = a; sr[t] = r;
  __syncthreads();
  for (int s = 128; s > 0; s >>= 1) {
    if (t < s) { sl[t] += sl[t + s]; sr[t] += sr[t + s]; }
    __syncthreads();
  }
  if (t == 0) {
    out[0] = sl[0] / (float)(B_ * N_);
    out[1] = sr[0] / (float)(B_ * N_);
  }
}

extern "C" void kernel_launch(void* const* d_in, const int* in_sizes, int n_in,
                              void* d_out, int out_size, void* d_ws, size_t ws_size,
                              hipStream_t stream) {
  (void)in_sizes; (void)n_in; (void)out_size; (void)ws_size;
  const float* srcDes = (const float*)d_in[0];
  const float* tgtDes = (const float*)d_in[1];
  const float* srcPts = (const float*)d_in[2];
  const float* tgtPts = (const float*)d_in[3];
  const float* tarUn  = (const float*)d_in[4];
  const int*   relax  = (const int*)d_in[5];

  float* refF = (float*)d_ws;                                  // B*N*C f32
  float* tarF = refF + (size_t)B_ * N_ * C_;                   // B*N*C f32
  int* mjArr = (int*)(tarF + (size_t)B_ * N_ * C_);            // B*N i32
  int* njArr = mjArr + B_ * N_;                                // B*N i32
  float* lossArr = (float*)(njArr + B_ * N_);                  // B*N f32
  float* recArr = lossArr + B_ * N_;                           // B*N f32
  float* out = (float*)d_out;

  sample_norm_kernel<<<2 * B_ * N_, 256, 0, stream>>>(
      srcDes, tgtDes, srcPts, tgtPts, refF, tarF);
  gemm_argmin_kernel<<<B_ * (N_ / 16), 256, 0, stream>>>(
      refF, tarF, tarUn, relax, mjArr, njArr);
  loss_kernel<<<(B_ * N_) / 16, 256, 0, stream>>>(
      refF, tarF, tarUn, mjArr, njArr, lossArr, recArr);
  reduce_kernel<<<1, 256, 0, stream>>>(lossArr, recArr, out);
}